// net3_56925496541248
// MI455X (gfx1250) — compile-verified
//
#include <hip/hip_runtime.h>
#include <math.h>

typedef __attribute__((ext_vector_type(2))) float v2f;
typedef __attribute__((ext_vector_type(4))) float v4f;
typedef __attribute__((ext_vector_type(8))) float v8f;

// ---------------------------------------------------------------------------
// 1) Hypernet GEMV: wv[r] = dot(gate[r,:], w) + b      (bandwidth bound:
//    1.93 GB of single-use gate traffic -> nontemporal b128 streaming loads,
//    80 KB w vector stays hot in cache and is reused by all 24112 rows)
// ---------------------------------------------------------------------------
__global__ void gemv_kernel(const float* __restrict__ gate,
                            const float* __restrict__ w,
                            const float* __restrict__ b,
                            float* __restrict__ wv, int ncols) {
  const int row = blockIdx.x;
  const v4f* g4 = (const v4f*)(gate + (size_t)row * (size_t)ncols);
  const v4f* w4 = (const v4f*)w;
  const int n4 = ncols >> 2;                       // 5000
  float acc0 = 0.f, acc1 = 0.f;
  int i = threadIdx.x;
  for (; i + 256 < n4; i += 512) {
    v4f ga = __builtin_nontemporal_load(&g4[i]);
    v4f gb = __builtin_nontemporal_load(&g4[i + 256]);
    v4f wa = w4[i];
    v4f wb = w4[i + 256];
    acc0 += ga.x * wa.x + ga.y * wa.y + ga.z * wa.z + ga.w * wa.w;
    acc1 += gb.x * wb.x + gb.y * wb.y + gb.z * wb.z + gb.w * wb.w;
  }
  if (i < n4) {
    v4f ga = __builtin_nontemporal_load(&g4[i]);
    v4f wa = w4[i];
    acc0 += ga.x * wa.x + ga.y * wa.y + ga.z * wa.z + ga.w * wa.w;
  }
  __shared__ float red[256];
  red[threadIdx.x] = acc0 + acc1;
  __syncthreads();
  for (int s = 128; s > 0; s >>= 1) {
    if (threadIdx.x < s) red[threadIdx.x] += red[threadIdx.x + s];
    __syncthreads();
  }
  if (threadIdx.x == 0) wv[row] = red[0] + b[0];
}

// ---------------------------------------------------------------------------
// 2) _norm_slice per segment (ddof=1 std), one block per segment
// ---------------------------------------------------------------------------
__global__ void norm_kernel(const float* __restrict__ wv, float* __restrict__ wn) {
  const int offs[5] = {0, 432, 5040, 23472, 24112};
  const int lo = offs[blockIdx.x], hi = offs[blockIdx.x + 1];
  const int n = hi - lo;
  __shared__ float s1[256], s2[256];
  float a = 0.f, q = 0.f;
  for (int i = threadIdx.x; i < n; i += 256) {
    float v = wv[lo + i];
    a += v; q += v * v;
  }
  s1[threadIdx.x] = a; s2[threadIdx.x] = q;
  __syncthreads();
  for (int s = 128; s > 0; s >>= 1) {
    if (threadIdx.x < s) { s1[threadIdx.x] += s1[threadIdx.x + s]; s2[threadIdx.x] += s2[threadIdx.x + s]; }
    __syncthreads();
  }
  const float mean = s1[0] / (float)n;
  const float var  = (s2[0] - (float)n * mean * mean) / (float)(n - 1);
  const float inv  = rsqrtf(var);
  for (int i = threadIdx.x; i < n; i += 256)
    wn[lo + i] = (wv[lo + i] - mean) * inv;
}

// ---------------------------------------------------------------------------
// 3) Pack weights into [K, Cout] GEMM-B layout, k ordered (ky, kx, cin)
//    B1: [28,16] (K=27 padded to 28), B2: [144,32], B3: [288,64], B4: [64,16]
// ---------------------------------------------------------------------------
__global__ void pack_weights_kernel(const float* __restrict__ wn,
                                    float* __restrict__ B1, float* __restrict__ B2,
                                    float* __restrict__ B3, float* __restrict__ B4) {
  int i = blockIdx.x * 256 + threadIdx.x;
  if (i < 448) {                       // B1 [28,16]
    int k = i >> 4, o = i & 15;
    float v = 0.f;
    if (k < 27) {
      int q = k / 3, c = k - q * 3, ky = q / 3, kx = q - ky * 3;
      v = wn[o * 27 + c * 9 + ky * 3 + kx];
    }
    B1[i] = v; return;
  }
  i -= 448;
  if (i < 4608) {                      // B2 [144,32]
    int k = i >> 5, o = i & 31;
    int q = k >> 4, c = k & 15, ky = q / 3, kx = q - ky * 3;
    B2[i] = wn[432 + o * 144 + c * 9 + ky * 3 + kx]; return;
  }
  i -= 4608;
  if (i < 18432) {                     // B3 [288,64]
    int k = i >> 6, o = i & 63;
    int q = k >> 5, c = k & 31, ky = q / 3, kx = q - ky * 3;
    B3[i] = wn[5040 + o * 288 + c * 9 + ky * 3 + kx]; return;
  }
  i -= 18432;
  if (i < 1024) {                      // B4 [64,16] = w4^T padded to 16 cols
    int k = i >> 4, n = i & 15;
    B4[i] = (n < 10) ? wn[23472 + n * 64 + k] : 0.f;
  }
}

// ---------------------------------------------------------------------------
// 4) Implicit-im2col conv as WMMA f32 16x16x4 GEMM, fully compile-time
//    specialized per layer (all div/mod constant-fold; K loop unrolls).
//    One wave = one 16x16 tile (16 output positions x 16 output channels).
//    dst layout: [N*Hout*Wout, Cout] (channel-fastest).
// ---------------------------------------------------------------------------
template <int Cin, int Hin, int Win, int Hout, int Wout, int Cout, int Kpad, bool NCHW>
__device__ __forceinline__ float conv_gather(const float* __restrict__ src,
                                             int k, int n, int oy, int ox) {
  constexpr int K = 9 * Cin;
  if (Kpad != K && k >= K) return 0.f;   // K padding (conv1: 27 -> 28)
  const int q  = k / Cin;                 // compile-time Cin -> shift/magic
  const int c  = k - q * Cin;
  const int ky = q / 3;
  const int kx = q - ky * 3;
  const int iy = oy * 2 - 1 + ky;
  const int ix = ox * 2 - 1 + kx;
  if (iy < 0 || iy >= Hin || ix < 0 || ix >= Win) return 0.f;
  if (NCHW)  // conv1 input x is NCHW
    return src[((n * Cin + c) * Hin + iy) * Win + ix];
  return src[(n * Hin * Win + iy * Win + ix) * Cin + c];
}

template <int Cin, int Hin, int Win, int Hout, int Wout, int Cout, int Kpad, bool NCHW>
__global__ void conv_wmma_kernel(const float* __restrict__ src,
                                 const float* __restrict__ Bw,
                                 float* __restrict__ dst, int total_tiles) {
  const int wave = blockIdx.x * (blockDim.x >> 5) + (threadIdx.x >> 5);
  if (wave >= total_tiles) return;       // wave-uniform exit -> EXEC all-1s at WMMA
  const int lane = threadIdx.x & 31;
  constexpr int tiles_n = Cout >> 4;
  const int tm = wave / tiles_n, tn = wave - tm * tiles_n;
  const int m0 = tm << 4, n0 = tn << 4;

  const int r = lane & 15;               // A/B: same 16 rows/cols in both halves
  const int khalf = (lane >> 4) * 2;     // K split across lane halves
  const int m = m0 + r;
  constexpr int hw = Hout * Wout;        // power of two -> shifts
  const int n  = m / hw;
  const int rem = m - n * hw;
  const int oy = rem / Wout;
  const int ox = rem - oy * Wout;
  const int col = n0 + r;

  v8f acc = {};
#pragma unroll 4
  for (int k = 0; k < Kpad; k += 4) {
    const int k0 = k + khalf;
    v2f a, bmat;
    a.x = conv_gather<Cin, Hin, Win, Hout, Wout, Cout, Kpad, NCHW>(src, k0,     n, oy, ox);
    a.y = conv_gather<Cin, Hin, Win, Hout, Wout, Cout, Kpad, NCHW>(src, k0 + 1, n, oy, ox);
    bmat.x = Bw[k0 * Cout + col];
    bmat.y = Bw[(k0 + 1) * Cout + col];
    acc = __builtin_amdgcn_wmma_f32_16x16x4_f32(false, a, false, bmat,
                                                (short)0, acc, false, false);
  }
  const int rowbase = (lane >> 4) * 8;
#pragma unroll
  for (int i = 0; i < 8; ++i)
    dst[(m0 + rowbase + i) * Cout + col] = acc[i];
}

// ---------------------------------------------------------------------------
// 5) BN batch stats: per-channel sum & sumsq over M rows of h[M,C]
//    stats layout: [0..C) = sum, [64..64+C) = sumsq  (zeroed before launch)
// ---------------------------------------------------------------------------
template <int C>
__global__ void bn_stats_kernel(const float* __restrict__ h, float* __restrict__ stats,
                                int total) {
  __shared__ float ssum[64], ssq[64];
  if (threadIdx.x < C) { ssum[threadIdx.x] = 0.f; ssq[threadIdx.x] = 0.f; }
  __syncthreads();
  const int T = gridDim.x * blockDim.x;  // T % C == 0 by construction
  const int g = blockIdx.x * blockDim.x + threadIdx.x;
  const int c = g & (C - 1);             // invariant across the stride loop
  float a = 0.f, q = 0.f;
  for (int idx = g; idx < total; idx += T) {
    float v = h[idx];
    a += v; q += v * v;
  }
  atomicAdd(&ssum[c], a);
  atomicAdd(&ssq[c], q);
  __syncthreads();
  if (threadIdx.x < C) {
    atomicAdd(&stats[threadIdx.x], ssum[threadIdx.x]);
    atomicAdd(&stats[64 + threadIdx.x], ssq[threadIdx.x]);
  }
}

// ---------------------------------------------------------------------------
// 6) BN(train) normalize + exact GELU, in place on h[M,C]
// ---------------------------------------------------------------------------
template <int C>
__global__ void bn_gelu_kernel(float* __restrict__ h, const float* __restrict__ stats,
                               const float* __restrict__ gamma, const float* __restrict__ beta,
                               int total, float invM) {
  const int i = blockIdx.x * 256 + threadIdx.x;
  if (i >= total) return;
  const int c = i & (C - 1);
  const float mean = stats[c] * invM;
  const float var  = stats[64 + c] * invM - mean * mean;
  const float xn = (h[i] - mean) * rsqrtf(var + 1e-5f);
  const float y = xn * gamma[c] + beta[c];
  h[i] = 0.5f * y * (1.f + erff(y * 0.70710678118654752f));
}

// ---------------------------------------------------------------------------
// 7) 4x4 average pool -> pooled[512,64]
// ---------------------------------------------------------------------------
__global__ void pool_kernel(const float* __restrict__ h3, float* __restrict__ pooled) {
  const int g = blockIdx.x * 256 + threadIdx.x;
  if (g >= 512 * 64) return;
  const int n = g >> 6, c = g & 63;
  float s = 0.f;
#pragma unroll
  for (int r = 0; r < 16; ++r) s += h3[(n * 16 + r) * 64 + c];
  pooled[g] = s * (1.f / 16.f);
}

// ---------------------------------------------------------------------------
// 8) Dense WMMA GEMM: C[M,N] = A[M,K] x B[K,N]  (final linear, K=64, N=16)
// ---------------------------------------------------------------------------
template <int K, int N>
__global__ void gemm_wmma_kernel(const float* __restrict__ A, const float* __restrict__ B,
                                 float* __restrict__ C, int total_tiles) {
  const int wave = blockIdx.x * (blockDim.x >> 5) + (threadIdx.x >> 5);
  if (wave >= total_tiles) return;
  const int lane = threadIdx.x & 31;
  const int m0 = wave << 4;
  const int r = lane & 15;
  const int khalf = (lane >> 4) * 2;
  v8f acc = {};
#pragma unroll
  for (int k = 0; k < K; k += 4) {
    const int k0 = k + khalf;
    v2f a, b;
    a.x = A[(m0 + r) * K + k0];
    a.y = A[(m0 + r) * K + k0 + 1];
    b.x = B[k0 * N + r];
    b.y = B[(k0 + 1) * N + r];
    acc = __builtin_amdgcn_wmma_f32_16x16x4_f32(false, a, false, b,
                                                (short)0, acc, false, false);
  }
  const int rowbase = (lane >> 4) * 8;
#pragma unroll
  for (int i = 0; i < 8; ++i)
    C[(m0 + rowbase + i) * N + r] = acc[i];
}

// ---------------------------------------------------------------------------
// 9) Slice padded [512,16] -> out [512,10]
// ---------------------------------------------------------------------------
__global__ void slice_kernel(const float* __restrict__ Cpad, float* __restrict__ out) {
  const int i = blockIdx.x * 256 + threadIdx.x;
  if (i >= 512 * 10) return;
  const int n = i / 10, o = i - n * 10;
  out[i] = Cpad[n * 16 + o];
}

// ---------------------------------------------------------------------------
extern "C" void kernel_launch(void* const* d_in, const int* in_sizes, int n_in,
                              void* d_out, int out_size, void* d_ws, size_t ws_size,
                              hipStream_t stream) {
  const float* x    = (const float*)d_in[0];  // [512,3,32,32]
  const float* gate = (const float*)d_in[1];  // [24112,20000]
  const float* w    = (const float*)d_in[2];  // [20000]
  const float* b    = (const float*)d_in[3];  // [1]
  const float* g1 = (const float*)d_in[4]; const float* be1 = (const float*)d_in[5];
  const float* g2 = (const float*)d_in[6]; const float* be2 = (const float*)d_in[7];
  const float* g3 = (const float*)d_in[8]; const float* be3 = (const float*)d_in[9];
  float* out = (float*)d_out;

  // workspace layout (floats), ~15.1 MB total
  float* ws = (float*)d_ws;
  float* wv     = ws;               // 24112
  float* wn     = wv + 24112;       // 24112
  float* B1     = wn + 24112;       // 448   [28,16]
  float* B2     = B1 + 448;         // 4608  [144,32]
  float* B3     = B2 + 4608;        // 18432 [288,64]
  float* B4     = B3 + 18432;       // 1024  [64,16]
  float* h1     = B4 + 1024;        // 131072*16  [pos, C]
  float* h2     = h1 + 131072 * 16; // 32768*32
  float* h3     = h2 + 32768 * 32;  // 8192*64
  float* pooled = h3 + 8192 * 64;   // 512*64
  float* Cpad   = pooled + 512 * 64;// 512*16
  float* stats  = Cpad + 512 * 16;  // 3 * 128

  hipMemsetAsync(stats, 0, 3 * 128 * sizeof(float), stream);

  // 1) hypernet GEMV (dominant: 1.93 GB streamed)
  gemv_kernel<<<24112, 256, 0, stream>>>(gate, w, b, wv, 20000);

  // 2) per-segment normalization
  norm_kernel<<<4, 256, 0, stream>>>(wv, wn);

  // 3) pack GEMM-B weight layouts
  pack_weights_kernel<<<(24512 + 255) / 256, 256, 0, stream>>>(wn, B1, B2, B3, B4);

  // 4) conv1: x[NCHW 512,3,32,32] -> h1[131072,16]; K=27 pad 28
  conv_wmma_kernel<3, 32, 32, 16, 16, 16, 28, true>
      <<<1024, 256, 0, stream>>>(x, B1, h1, 8192);
  bn_stats_kernel<16><<<512, 256, 0, stream>>>(h1, stats, 131072 * 16);
  bn_gelu_kernel<16><<<(131072 * 16 + 255) / 256, 256, 0, stream>>>(
      h1, stats, g1, be1, 131072 * 16, 1.f / 131072.f);

  // 5) conv2: h1 -> h2[32768,32]; K=144
  conv_wmma_kernel<16, 16, 16, 8, 8, 32, 144, false>
      <<<512, 256, 0, stream>>>(h1, B2, h2, 4096);
  bn_stats_kernel<32><<<512, 256, 0, stream>>>(h2, stats + 128, 32768 * 32);
  bn_gelu_kernel<32><<<(32768 * 32 + 255) / 256, 256, 0, stream>>>(
      h2, stats + 128, g2, be2, 32768 * 32, 1.f / 32768.f);

  // 6) conv3: h2 -> h3[8192,64]; K=288
  conv_wmma_kernel<32, 8, 8, 4, 4, 64, 288, false>
      <<<256, 256, 0, stream>>>(h2, B3, h3, 2048);
  bn_stats_kernel<64><<<512, 256, 0, stream>>>(h3, stats + 256, 8192 * 64);
  bn_gelu_kernel<64><<<(8192 * 64 + 255) / 256, 256, 0, stream>>>(
      h3, stats + 256, g3, be3, 8192 * 64, 1.f / 8192.f);

  // 7) pool -> pooled[512,64]
  pool_kernel<<<(512 * 64 + 255) / 256, 256, 0, stream>>>(h3, pooled);

  // 8) final linear via WMMA: pooled[512,64] x B4[64,16] -> Cpad[512,16]
  gemm_wmma_kernel<64, 16><<<4, 256, 0, stream>>>(pooled, B4, Cpad, 32);

  // 9) slice to [512,10]
  slice_kernel<<<(512 * 10 + 255) / 256, 256, 0, stream>>>(Cpad, out);
}